// MPNNBlockMultSingleEtype_35192962023430
// MI455X (gfx1250) — compile-verified
//
#include <hip/hip_runtime.h>
#include <hip/hip_bf16.h>

typedef __attribute__((ext_vector_type(2))) float v2f;
typedef __attribute__((ext_vector_type(8))) float v8f;

#define HID 128   // HIDDEN == OUT == 128

// ---------------------------------------------------------------------------
// Kernel 1: zero sums[N*128] + deg[N] (contiguous region in d_ws)
// ---------------------------------------------------------------------------
__global__ __launch_bounds__(256) void zero_ws_kernel(float* __restrict__ p, int count) {
    int i = blockIdx.x * blockDim.x + threadIdx.x;
    if (i < count) p[i] = 0.0f;
}

// ---------------------------------------------------------------------------
// Kernel 2: node-level dual GEMM via V_WMMA_F32_16X16X4_F32.
//   msg   = h @ We^T          (N x 128) -> workspace
//   selfp = h @ Ws^T          (N x 128) -> d_out (pre-activation staging)
// One wave per 16-row tile of h. K = 128 -> 32 wmma steps per 16x16 tile.
// A fragment (16x4 f32, ISA 7.12.2): lane<16 holds M=lane, K={k0,k0+1};
//                                    lane>=16 holds M=lane-16, K={k0+2,k0+3}.
// B fragment (4x16): mirrored lane-half split over K; B[k][n] = W[n*HID + k].
// D tile: VGPR i -> (M = i + 8*laneHalf, N = lane&15).
// ---------------------------------------------------------------------------
__global__ __launch_bounds__(256) void node_gemm_wmma_kernel(
    const float* __restrict__ h,
    const float* __restrict__ We,
    const float* __restrict__ Ws,
    float* __restrict__ msg,
    float* __restrict__ selfp,
    int n_nodes, int ntiles)
{
    int wave = (blockIdx.x * blockDim.x + threadIdx.x) >> 5;
    if (wave >= ntiles) return;              // whole-wave guard: EXEC stays all-1s
    int lane = threadIdx.x & 31;
    int half = lane >> 4;                    // 0: lanes 0-15, 1: lanes 16-31
    int l    = lane & 15;

    int m0 = wave * 16;
    int arow = m0 + l;
    if (arow >= n_nodes) arow = n_nodes - 1; // clamp (keeps EXEC full); unused rows masked on store

    // ---- load all A fragments for this 16-row strip (K = 128) ----
    const float* hrow = h + (size_t)arow * HID + half * 2;
    v2f a[32];
#pragma unroll
    for (int kk = 0; kk < 32; ++kk) {
        a[kk].x = hrow[kk * 4 + 0];
        a[kk].y = hrow[kk * 4 + 1];
    }

    // ---- 8 column tiles of 16 outputs each ----
    for (int ct = 0; ct < 8; ++ct) {
        int n0 = ct * 16;
        const float* wce = We + (size_t)(n0 + l) * HID + half * 2;  // column (n0+l) of We^T
        const float* wcs = Ws + (size_t)(n0 + l) * HID + half * 2;

        v8f accE = {};
        v8f accS = {};
#pragma unroll
        for (int kk = 0; kk < 32; ++kk) {
            v2f be, bs;
            be.x = wce[kk * 4 + 0];
            be.y = wce[kk * 4 + 1];
            bs.x = wcs[kk * 4 + 0];
            bs.y = wcs[kk * 4 + 1];
            accE = __builtin_amdgcn_wmma_f32_16x16x4_f32(
                false, a[kk], false, be, (short)0, accE, false, false);
            accS = __builtin_amdgcn_wmma_f32_16x16x4_f32(
                false, a[kk], false, bs, (short)0, accS, false, false);
        }

        // ---- store D tiles ----
#pragma unroll
        for (int i = 0; i < 8; ++i) {
            int row = m0 + half * 8 + i;
            if (row < n_nodes) {
                size_t off = (size_t)row * HID + (n0 + l);
                msg[off]   = accE[i];
                selfp[off] = accS[i];
            }
        }
    }
}

// ---------------------------------------------------------------------------
// Kernel 3: edge gather + scatter-add. One wave per edge (grid-stride).
// Lane j moves 4 consecutive floats (b128 load, 4x global_atomic_add_f32).
// ---------------------------------------------------------------------------
__global__ __launch_bounds__(256) void edge_scatter_kernel(
    const float* __restrict__ msg,
    const int*   __restrict__ src,
    const int*   __restrict__ dst,
    float* __restrict__ sums,
    float* __restrict__ deg,
    int n_edges, int wave_stride)
{
    int wave = (blockIdx.x * blockDim.x + threadIdx.x) >> 5;
    int lane = threadIdx.x & 31;

    for (int e = wave; e < n_edges; e += wave_stride) {
        int s = src[e];
        int d = dst[e];

        // prefetch next edge's source row into cache (global_prefetch_b8)
        int en = e + wave_stride;
        if (en < n_edges) {
            int sn = src[en];
            __builtin_prefetch(msg + (size_t)sn * HID + lane * 4, 0, 1);
        }

        const float4 v = ((const float4*)(msg + (size_t)s * HID))[lane];
        float* base = sums + (size_t)d * HID + lane * 4;
        atomicAdd(base + 0, v.x);
        atomicAdd(base + 1, v.y);
        atomicAdd(base + 2, v.z);
        atomicAdd(base + 3, v.w);
        if (lane == 0) atomicAdd(deg + d, 1.0f);
    }
}

// ---------------------------------------------------------------------------
// Kernel 4: out = relu(self_pre + b_self + sums / max(deg,1)), float4-wide.
// ---------------------------------------------------------------------------
__global__ __launch_bounds__(256) void finalize_kernel(
    float* __restrict__ out,
    const float* __restrict__ sums,
    const float* __restrict__ deg,
    const float* __restrict__ bias,
    int total4)
{
    int t = blockIdx.x * blockDim.x + threadIdx.x;
    if (t >= total4) return;
    int e0 = t * 4;
    int n  = e0 >> 7;          // / 128
    int c  = e0 & (HID - 1);   // % 128 (multiple of 4)

    float4 sp = ((const float4*)out)[t];
    float4 sm = ((const float4*)sums)[t];
    float4 bv = *(const float4*)(bias + c);
    float inv = 1.0f / fmaxf(deg[n], 1.0f);

    float4 r;
    r.x = fmaxf(sp.x + bv.x + sm.x * inv, 0.0f);
    r.y = fmaxf(sp.y + bv.y + sm.y * inv, 0.0f);
    r.z = fmaxf(sp.z + bv.z + sm.z * inv, 0.0f);
    r.w = fmaxf(sp.w + bv.w + sm.w * inv, 0.0f);
    ((float4*)out)[t] = r;
}

// ---------------------------------------------------------------------------
extern "C" void kernel_launch(void* const* d_in, const int* in_sizes, int n_in,
                              void* d_out, int out_size, void* d_ws, size_t ws_size,
                              hipStream_t stream) {
    const float* h    = (const float*)d_in[0];
    const int*   src  = (const int*)d_in[1];
    const int*   dst  = (const int*)d_in[2];
    const float* We   = (const float*)d_in[3];
    const float* Ws   = (const float*)d_in[4];
    const float* bias = (const float*)d_in[5];
    float* out = (float*)d_out;

    const int N = in_sizes[0] / HID;   // 50000
    const int E = in_sizes[1];         // 800000

    // workspace layout
    float* msg  = (float*)d_ws;                    // N*128 floats
    float* sums = msg  + (size_t)N * HID;          // N*128 floats
    float* deg  = sums + (size_t)N * HID;          // N floats

    // 1) zero sums + deg (contiguous)
    {
        int zc = N * HID + N;
        int blocks = (zc + 255) / 256;
        zero_ws_kernel<<<blocks, 256, 0, stream>>>(sums, zc);
    }

    // 2) dual GEMM (msg = h@We^T -> ws ; self_pre = h@Ws^T -> d_out)
    {
        int ntiles = (N + 15) / 16;                // 3125
        int blocks = (ntiles + 7) / 8;             // 8 waves / block
        node_gemm_wmma_kernel<<<blocks, 256, 0, stream>>>(h, We, Ws, msg, out, N, ntiles);
    }

    // 3) edge gather/scatter
    {
        const int blocks = 4096;                   // 32768 waves, grid-stride
        const int wave_stride = blocks * (256 / 32);
        edge_scatter_kernel<<<blocks, 256, 0, stream>>>(msg, src, dst, sums, deg, E, wave_stride);
    }

    // 4) finalize
    {
        int total4 = (N * HID) / 4;
        int blocks = (total4 + 255) / 256;
        finalize_kernel<<<blocks, 256, 0, stream>>>(out, sums, deg, bias, total4);
    }
}